// PointActorTFNTraj_52295521796464
// MI455X (gfx1250) — compile-verified
//
#include <hip/hip_runtime.h>
#include <hip/hip_bf16.h>

// Problem constants (from reference)
#define B_      64
#define N_PER_  4096
#define N_TOOL_ 1000
#define D_      128
#define T_      50
#define OUT3_   150        // 3*T
#define SCALE_  50.0f
#define TILES_  63         // ceil(1000/16)
#define LDW     132        // 128 + 4 pad floats -> conflict-free LDS rows

typedef __attribute__((ext_vector_type(2))) float        v2f;
typedef __attribute__((ext_vector_type(8))) float        v8f;
typedef __attribute__((ext_vector_type(4))) unsigned int u32x4;
typedef __attribute__((ext_vector_type(8))) int          i32x8;
typedef __attribute__((ext_vector_type(4))) int          i32x4;

// The TDM builtin has different arity on the two known gfx1250 toolchains.
// The gfx1250_TDM.h header ships only with the 6-arg (therock-10.0) headers.
#if __has_include(<hip/amd_detail/amd_gfx1250_TDM.h>)
#define TDM_6ARG 1
#else
#define TDM_6ARG 0
#endif

// ---------------------------------------------------------------------------
// One 16x16 output chunk of a [16 x 128] * [128 x ncols] GEMM using
// V_WMMA_F32_16X16X4_F32 (K swept in steps of 4).
// A layout (ISA 7.12.2): lane&15 = M row; VGPR0/1 hold K = 2*(lane>>4) + {0,1}.
// B layout: row-striped across lanes; lane&15 = N col, K mirrors A's grouping.
// C/D layout: 8 VGPRs, row M = v + 8*(lane>>4), col N = lane&15.
// ---------------------------------------------------------------------------
__device__ __forceinline__ v8f wmma_rowchunk(const float (*__restrict__ inb)[LDW],
                                             const float* __restrict__ W, int ldw,
                                             int n, int ncols, int lane) {
  const int lo = lane & 15;
  const int hi = lane >> 4;
  const int m  = lo;
  const bool nok = (n < ncols);
  v8f acc = {0.f, 0.f, 0.f, 0.f, 0.f, 0.f, 0.f, 0.f};
#pragma unroll
  for (int ks = 0; ks < D_ / 4; ++ks) {
    const int k = ks * 4 + 2 * hi;
    v2f a, b;
    a.x = inb[m][k];
    a.y = inb[m][k + 1];
    b.x = nok ? W[(size_t)k * ldw + n] : 0.f;
    b.y = nok ? W[(size_t)(k + 1) * ldw + n] : 0.f;
    acc = __builtin_amdgcn_wmma_f32_16x16x4_f32(false, a, false, b,
                                                (short)0, acc, false, false);
  }
  return acc;
}

// ---------------------------------------------------------------------------
// Kernel 1: TDM-gather 16 tool rows into LDS -> 3-layer MLP (fp32 WMMA) ->
// write flows directly into the "before_svd" half of d_out, (B,T,N_TOOL,3).
// One wave32 per 16-row tile. grid = B * 63 = 4032 blocks of 32 threads.
// ---------------------------------------------------------------------------
__global__ void __launch_bounds__(32)
mlp_kernel(const float* __restrict__ x_feat, const int* __restrict__ tool_idx,
           const float* __restrict__ W1, const float* __restrict__ b1,
           const float* __restrict__ W2, const float* __restrict__ b2,
           const float* __restrict__ W3, const float* __restrict__ b3,
           float* __restrict__ out2) {
  __shared__ float bufA[16][LDW];
  __shared__ float bufB[16][LDW];

  const int lane    = threadIdx.x;       // wave32
  const int tile_id = blockIdx.x;
  const int b       = tile_id / TILES_;
  const int tile    = tile_id % TILES_;
  const int lo      = lane & 15;
  const int hi      = lane >> 4;

  // Token weight prefetch (global_prefetch_b8); weights are L2-resident.
  __builtin_prefetch(W1 + lane * 32, 0, 0);
  __builtin_prefetch(W2 + lane * 32, 0, 0);
  __builtin_prefetch(W3 + lane * 32, 0, 0);

  // -------- TDM gather: 16 scattered rows (128 f32) of x_feat -> bufA -------
  // D# gather mode (CDNA5 ISA): 16-bit row indices, 2D tile.
  // Pad feature recreates the 132-float LDS row stride (pad 4 DWORDs every
  // 128 DWORDs) so WMMA A-fragment reads stay bank-conflict free.
  const int* tidx = tool_idx + (size_t)b * N_TOOL_;
  unsigned int ridx[16];
#pragma unroll
  for (int r = 0; r < 16; ++r) {
    int nt = tile * 16 + r;
    ridx[r] = (unsigned int)tidx[nt < N_TOOL_ ? nt : (N_TOOL_ - 1)] & 0xffffu;
  }

  const unsigned int  lds_off = (unsigned int)(uintptr_t)&bufA[0][0]; // LDS byte offset
  const unsigned long gaddr   = (unsigned long)(uintptr_t)(x_feat + (size_t)b * N_PER_ * D_);

  u32x4 g0;
  g0[0] = 1u | (0u << 30) | (1u << 31);            // count=1, 16-bit idx, gather_mode=1
  g0[1] = lds_off;                                 // lds_addr
  g0[2] = (unsigned int)(gaddr & 0xffffffffu);     // global_addr[31:0]
  g0[3] = (unsigned int)((gaddr >> 32) & 0x01ffffffu) | (2u << 30); // addr[56:32], type=2

  i32x8 g1;
  g1[0] = (int)((2u << 16)        // data_size = 4 bytes
              | (1u << 20)        // pad_enable
              | (6u << 22)        // pad_interval code 6 = every 128 DWORDs
              | (3u << 25));      // pad_amount  code 3 = 4 DWORDs
  g1[1] = (int)(128u << 16);      // tensor_dim0 = 128 (bits 79:48)
  g1[2] = (int)((unsigned)(N_PER_ & 0xffff) << 16); // tensor_dim1 lo16 (bits 111:80)
  g1[3] = (int)(128u << 16);      // tensor_dim1 hi16 = 0 | tile_dim0 = 128
  g1[4] = 16;                     // tile_dim1 = #valid gather indices
  g1[5] = 128;                    // tensor_dim0_stride = 128 elements
  g1[6] = 0;
  g1[7] = 0;

  i32x4 g2, g3;
#pragma unroll
  for (int p = 0; p < 4; ++p) {
    g2[p] = (int)(ridx[2 * p]     | (ridx[2 * p + 1] << 16));   // rows 0..7
    g3[p] = (int)(ridx[8 + 2 * p] | (ridx[9 + 2 * p] << 16));   // rows 8..15
  }

#if TDM_6ARG
  i32x8 gpad = {0, 0, 0, 0, 0, 0, 0, 0};
  __builtin_amdgcn_tensor_load_to_lds(g0, g1, g2, g3, gpad, 0);
#else
  __builtin_amdgcn_tensor_load_to_lds(g0, g1, g2, g3, 0);
#endif
  __builtin_amdgcn_s_wait_tensorcnt(0);
  __syncthreads();

  // -------- layer 1: bufB = relu(bufA @ W1 + b1) --------
  for (int nc = 0; nc < 8; ++nc) {
    int n = nc * 16 + lo;
    v8f acc = wmma_rowchunk(bufA, W1, D_, n, D_, lane);
    float bias = b1[n];
#pragma unroll
    for (int v = 0; v < 8; ++v)
      bufB[v + 8 * hi][n] = fmaxf(acc[v] + bias, 0.f);
  }
  __syncthreads();

  // -------- layer 2: bufA = relu(bufB @ W2 + b2) --------
  for (int nc = 0; nc < 8; ++nc) {
    int n = nc * 16 + lo;
    v8f acc = wmma_rowchunk(bufB, W2, D_, n, D_, lane);
    float bias = b2[n];
#pragma unroll
    for (int v = 0; v < 8; ++v)
      bufA[v + 8 * hi][n] = fmaxf(acc[v] + bias, 0.f);
  }
  __syncthreads();

  // -------- layer 3: flows = bufA @ W3 + b3, scatter to (B,T,N_TOOL,3) -----
  for (int nc = 0; nc < 10; ++nc) {                // 160 cols padded (150 real)
    int n = nc * 16 + lo;
    v8f acc = wmma_rowchunk(bufA, W3, OUT3_, n, OUT3_, lane);
    float bias = (n < OUT3_) ? b3[n] : 0.f;
#pragma unroll
    for (int v = 0; v < 8; ++v) {
      int mrow = v + 8 * hi;
      int nt   = tile * 16 + mrow;
      if (nt < N_TOOL_ && n < OUT3_) {
        int t = n / 3;
        int j = n - 3 * t;
        out2[(((size_t)b * T_ + t) * N_TOOL_ + nt) * 3 + j] = acc[v] + bias;
      }
    }
  }
}

// ---------------------------------------------------------------------------
// 3x3 Kabsch rotation from cross-covariance H (single thread).
// SVD via Jacobi eigen-decomposition of H^T H; det-sign fix per reference Dm.
// ---------------------------------------------------------------------------
__device__ __forceinline__ float det3(const float M[3][3]) {
  return M[0][0] * (M[1][1] * M[2][2] - M[1][2] * M[2][1])
       - M[0][1] * (M[1][0] * M[2][2] - M[1][2] * M[2][0])
       + M[0][2] * (M[1][0] * M[2][1] - M[1][1] * M[2][0]);
}

__device__ void jrot(float S[3][3], float V[3][3], int p, int q) {
  float apq = S[p][q];
  if (fabsf(apq) < 1e-20f) return;
  float tau = (S[q][q] - S[p][p]) / (2.f * apq);
  float t   = ((tau >= 0.f) ? 1.f : -1.f) / (fabsf(tau) + sqrtf(1.f + tau * tau));
  float c   = 1.f / sqrtf(1.f + t * t);
  float s   = t * c;
#pragma unroll
  for (int k = 0; k < 3; ++k) {
    float skp = S[k][p], skq = S[k][q];
    S[k][p] = c * skp - s * skq;
    S[k][q] = s * skp + c * skq;
  }
#pragma unroll
  for (int k = 0; k < 3; ++k) {
    float spk = S[p][k], sqk = S[q][k];
    S[p][k] = c * spk - s * sqk;
    S[q][k] = s * spk + c * sqk;
  }
#pragma unroll
  for (int k = 0; k < 3; ++k) {
    float vkp = V[k][p], vkq = V[k][q];
    V[k][p] = c * vkp - s * vkq;
    V[k][q] = s * vkp + c * vkq;
  }
}

__device__ void kabsch_R(const float H[3][3], float R[3][3]) {
  float S[3][3], V[3][3];
#pragma unroll
  for (int i = 0; i < 3; ++i)
#pragma unroll
    for (int j = 0; j < 3; ++j) {
      S[i][j] = H[0][i] * H[0][j] + H[1][i] * H[1][j] + H[2][i] * H[2][j];
      V[i][j] = (i == j) ? 1.f : 0.f;
    }
  for (int sweep = 0; sweep < 6; ++sweep) {
    jrot(S, V, 0, 1);
    jrot(S, V, 0, 2);
    jrot(S, V, 1, 2);
  }
  float lam[3] = {S[0][0], S[1][1], S[2][2]};
  // sort eigenpairs descending (permute V columns)
#pragma unroll
  for (int a = 0; a < 2; ++a)
    for (int p = 0; p < 2 - a; ++p)
      if (lam[p] < lam[p + 1]) {
        float tl = lam[p]; lam[p] = lam[p + 1]; lam[p + 1] = tl;
#pragma unroll
        for (int k = 0; k < 3; ++k) {
          float tv = V[k][p]; V[k][p] = V[k][p + 1]; V[k][p + 1] = tv;
        }
      }
  float U[3][3], sg[3];
#pragma unroll
  for (int c = 0; c < 3; ++c) {
    sg[c] = sqrtf(fmaxf(lam[c], 0.f));
    float u0 = H[0][0] * V[0][c] + H[0][1] * V[1][c] + H[0][2] * V[2][c];
    float u1 = H[1][0] * V[0][c] + H[1][1] * V[1][c] + H[1][2] * V[2][c];
    float u2 = H[2][0] * V[0][c] + H[2][1] * V[1][c] + H[2][2] * V[2][c];
    float inv = (sg[c] > 1e-7f) ? (1.f / sg[c]) : 0.f;
    U[0][c] = u0 * inv; U[1][c] = u1 * inv; U[2][c] = u2 * inv;
  }
  if (sg[2] <= 1e-7f) {  // rank-deficient: complete U with cross product
    float cx = U[1][0] * U[2][1] - U[2][0] * U[1][1];
    float cy = U[2][0] * U[0][1] - U[0][0] * U[2][1];
    float cz = U[0][0] * U[1][1] - U[1][0] * U[0][1];
    float nrm = sqrtf(cx * cx + cy * cy + cz * cz);
    float inv = (nrm > 1e-12f) ? (1.f / nrm) : 0.f;
    U[0][2] = cx * inv; U[1][2] = cy * inv; U[2][2] = cz * inv;
  }
  float d = (det3(U) * det3(V) < 0.f) ? -1.f : 1.f;   // sign(det(U @ Vt))
#pragma unroll
  for (int i = 0; i < 3; ++i)
#pragma unroll
    for (int j = 0; j < 3; ++j)
      R[i][j] = U[i][0] * V[j][0] + U[i][1] * V[j][1] + d * U[i][2] * V[j][2];
}

// ---------------------------------------------------------------------------
// Kernel 2: per-batch serial Kabsch scan. One block per batch, 256 threads.
// Point state lives in registers; reductions via shfl + ds_add_f32 atomics.
// ---------------------------------------------------------------------------
__global__ void __launch_bounds__(256)
kabsch_kernel(const float* __restrict__ pos, const int* __restrict__ tool_idx,
              const float* __restrict__ flows, float* __restrict__ out1) {
  const int b    = blockIdx.x;
  const int tid  = threadIdx.x;
  const int lane = tid & 31;

  __shared__ float red[15];   // sum x (3), sum y (3), sum x_i*y_j (9)
  __shared__ float RM[15];    // R (9), mu_x (3), mu_y (3)

  constexpr int NPT = (N_TOOL_ + 255) / 256;  // 4
  float x[NPT][3];
  int   nt[NPT];
#pragma unroll
  for (int i = 0; i < NPT; ++i) {
    int p = tid + i * 256;
    nt[i] = p;
    if (p < N_TOOL_) {
      int idx = tool_idx[(size_t)b * N_TOOL_ + p];
      size_t base = ((size_t)b * N_PER_ + idx) * 3;
      x[i][0] = pos[base + 0] * SCALE_;
      x[i][1] = pos[base + 1] * SCALE_;
      x[i][2] = pos[base + 2] * SCALE_;
    } else {
      x[i][0] = x[i][1] = x[i][2] = 0.f;
    }
  }

  for (int t = 0; t < T_; ++t) {
    if (tid < 15) red[tid] = 0.f;
    __syncthreads();

    float f[NPT][3];
    float s[15];
#pragma unroll
    for (int k = 0; k < 15; ++k) s[k] = 0.f;
#pragma unroll
    for (int i = 0; i < NPT; ++i) {
      if (nt[i] < N_TOOL_) {
        size_t base = (((size_t)b * T_ + t) * N_TOOL_ + nt[i]) * 3;
        f[i][0] = flows[base + 0];
        f[i][1] = flows[base + 1];
        f[i][2] = flows[base + 2];
        float y0 = x[i][0] + f[i][0];
        float y1 = x[i][1] + f[i][1];
        float y2 = x[i][2] + f[i][2];
        s[0] += x[i][0]; s[1] += x[i][1]; s[2] += x[i][2];
        s[3] += y0;      s[4] += y1;      s[5] += y2;
        s[6]  += x[i][0] * y0; s[7]  += x[i][0] * y1; s[8]  += x[i][0] * y2;
        s[9]  += x[i][1] * y0; s[10] += x[i][1] * y1; s[11] += x[i][1] * y2;
        s[12] += x[i][2] * y0; s[13] += x[i][2] * y1; s[14] += x[i][2] * y2;
      }
    }
    // wave32 tree reduction, then one ds_add_f32 per value per wave
#pragma unroll
    for (int off = 16; off >= 1; off >>= 1)
#pragma unroll
      for (int k = 0; k < 15; ++k)
        s[k] += __shfl_down(s[k], off, 32);
    if (lane == 0)
#pragma unroll
      for (int k = 0; k < 15; ++k) atomicAdd(&red[k], s[k]);
    __syncthreads();

    if (tid == 0) {
      const float invN = 1.f / (float)N_TOOL_;
      float mux[3] = {red[0] * invN, red[1] * invN, red[2] * invN};
      float muy[3] = {red[3] * invN, red[4] * invN, red[5] * invN};
      float H[3][3];
#pragma unroll
      for (int i = 0; i < 3; ++i)
#pragma unroll
        for (int j = 0; j < 3; ++j)
          H[i][j] = red[6 + 3 * i + j] - (float)N_TOOL_ * mux[i] * muy[j];
      float R[3][3];
      kabsch_R(H, R);
#pragma unroll
      for (int i = 0; i < 3; ++i)
#pragma unroll
        for (int j = 0; j < 3; ++j) RM[3 * i + j] = R[i][j];
      RM[9] = mux[0]; RM[10] = mux[1]; RM[11] = mux[2];
      RM[12] = muy[0]; RM[13] = muy[1]; RM[14] = muy[2];
    }
    __syncthreads();

    float R00 = RM[0], R01 = RM[1], R02 = RM[2];
    float R10 = RM[3], R11 = RM[4], R12 = RM[5];
    float R20 = RM[6], R21 = RM[7], R22 = RM[8];
    float mx0 = RM[9], mx1 = RM[10], mx2 = RM[11];
    float my0 = RM[12], my1 = RM[13], my2 = RM[14];
#pragma unroll
    for (int i = 0; i < NPT; ++i) {
      if (nt[i] < N_TOOL_) {
        float xc0 = x[i][0] - mx0, xc1 = x[i][1] - mx1, xc2 = x[i][2] - mx2;
        float n0 = xc0 * R00 + xc1 * R10 + xc2 * R20 + my0;
        float n1 = xc0 * R01 + xc1 * R11 + xc2 * R21 + my1;
        float n2 = xc0 * R02 + xc1 * R12 + xc2 * R22 + my2;
        size_t base = (((size_t)b * T_ + t) * N_TOOL_ + nt[i]) * 3;
        out1[base + 0] = n0 - x[i][0];
        out1[base + 1] = n1 - x[i][1];
        out1[base + 2] = n2 - x[i][2];
        x[i][0] = n0; x[i][1] = n1; x[i][2] = n2;
      }
    }
    __syncthreads();
  }
}

// ---------------------------------------------------------------------------
extern "C" void kernel_launch(void* const* d_in, const int* in_sizes, int n_in,
                              void* d_out, int out_size, void* d_ws, size_t ws_size,
                              hipStream_t stream) {
  const float* x_feat   = (const float*)d_in[0];
  const float* pos      = (const float*)d_in[1];
  const int*   tool_idx = (const int*)d_in[2];
  const float* W1 = (const float*)d_in[3];
  const float* b1 = (const float*)d_in[4];
  const float* W2 = (const float*)d_in[5];
  const float* b2 = (const float*)d_in[6];
  const float* W3 = (const float*)d_in[7];
  const float* b3 = (const float*)d_in[8];

  float* out1 = (float*)d_out;                                  // flow_traj
  float* out2 = out1 + (size_t)B_ * T_ * N_TOOL_ * 3;           // before_svd

  // Phase 1: TDM gather + MLP (fp32 WMMA), writes before_svd/flows.
  mlp_kernel<<<B_ * TILES_, 32, 0, stream>>>(x_feat, tool_idx,
                                             W1, b1, W2, b2, W3, b3, out2);
  // Phase 2: serial per-batch Kabsch scan, writes flow_traj.
  kabsch_kernel<<<B_, 256, 0, stream>>>(pos, tool_idx, out2, out1);
}